// ReadGate_77068893160216
// MI455X (gfx1250) — compile-verified
//
#include <hip/hip_runtime.h>
#include <hip/hip_bf16.h>

#define HD 64          // HIDDEN_DIM == VOCAB_SIZE
#define MROWS 2048
#define TILE_ROWS 128
#define NT (MROWS / TILE_ROWS)   // 16 tiles per row b
#define BLOCK 256
#define WAVES 8

typedef __attribute__((ext_vector_type(2))) float v2f;
typedef __attribute__((ext_vector_type(8))) float v8f;
typedef __attribute__((ext_vector_type(4))) int   v4i;

typedef __attribute__((address_space(1))) v4i gv4i;   // global (device) int4
typedef __attribute__((address_space(3))) v4i lv4i;   // LDS int4

#if __has_builtin(__builtin_amdgcn_global_load_async_to_lds_b128)
#define HAVE_ASYNC 1
#endif

__device__ __forceinline__ void wait_async_zero() {
#ifdef HAVE_ASYNC
#if __has_builtin(__builtin_amdgcn_s_wait_asynccnt)
    __builtin_amdgcn_s_wait_asynccnt(0);
#else
    asm volatile("s_wait_asynccnt 0" ::: "memory");
#endif
#endif
}

__global__ __launch_bounds__(BLOCK) void attn_pool_kernel(
    const int* __restrict__ query, const float* __restrict__ mem_,
    const float* __restrict__ emb, const float* __restrict__ qW,
    const float* __restrict__ qb, const float* __restrict__ oW,
    const float* __restrict__ ob, float* __restrict__ out)
{
    __shared__ float lds_tile[2][TILE_ROWS * HD];   // 2 x 32 KB double buffer
    __shared__ float lds_q[HD];
    __shared__ float lds_ms[WAVES][2];              // per-wave (max, sum)
    __shared__ float lds_pool[WAVES][HD];
    __shared__ float lds_pooled[HD];

    const int b    = blockIdx.x;
    const int tid  = threadIdx.x;
    const int lane = tid & 31;
    const int wave = tid >> 5;
    const int hi   = lane >> 4;      // which half of the wave
    const float* memb = mem_ + (size_t)b * MROWS * HD;

    // ---- q = emb[query[b]] @ qW^T + qb  (64 threads, 64 FMAs each) ----
    if (tid < HD) {
        const int qi = query[b];
        const float* er = emb + qi * HD;
        const float* wr = qW + tid * HD;
        float acc = qb[tid];
        #pragma unroll 8
        for (int k = 0; k < HD; ++k) acc = fmaf(er[k], wr[k], acc);
        lds_q[tid] = acc;
    }
    __syncthreads();

    // ---- preload WMMA B fragments: q broadcast into all 16 columns ----
    // f32 16x16x4 B frag: 2 VGPRs/lane; lanes 0-15 hold k=0/1, lanes 16-31 k=2/3
    v2f bfrag[16];
    #pragma unroll
    for (int c = 0; c < 16; ++c) {
        bfrag[c].x = lds_q[4 * c + 2 * hi];
        bfrag[c].y = lds_q[4 * c + 2 * hi + 1];
    }

    // ---- tile loader: 128 rows x 64 cols fp32 = 32 KB, 8 x b128 per thread ----
    auto issue_tile = [&](int t, int buf) {
        const float* src = memb + (size_t)t * TILE_ROWS * HD;
        #pragma unroll
        for (int i = 0; i < 8; ++i) {
            const int idx = i * BLOCK + tid;           // float4 index
#ifdef HAVE_ASYNC
            __builtin_amdgcn_global_load_async_to_lds_b128(
                (gv4i*)(src + idx * 4),
                (lv4i*)&lds_tile[buf][idx * 4],
                0, 0);
#else
            const float4 v = *(const float4*)(src + idx * 4);
            *(float4*)&lds_tile[buf][idx * 4] = v;
#endif
        }
    };

    // ---- online-softmax streaming pass over memory[b] ----
    float mW = -__builtin_inff();
    float sW = 0.f;
    float px = 0.f, py = 0.f;   // pooled dims 2*lane, 2*lane+1

    issue_tile(0, 0);
    for (int t = 0; t < NT; ++t) {
        const int cur = t & 1;
        wait_async_zero();
        __syncthreads();                 // tile t fully in LDS; buf cur^1 free
        if (t + 1 < NT) issue_tile(t + 1, cur ^ 1);

        const float* rowbase = &lds_tile[cur][0] + wave * 16 * HD;  // this wave's 16 rows
        const int r = lane & 15;
        const float* arow = rowbase + r * HD;

        // sims for 16 rows via 16 x V_WMMA_F32_16X16X4_F32 (K = 64)
        v8f acc = {0.f, 0.f, 0.f, 0.f, 0.f, 0.f, 0.f, 0.f};
        #pragma unroll
        for (int c = 0; c < 16; ++c) {
            v2f a;
            a.x = arow[4 * c + 2 * hi];
            a.y = arow[4 * c + 2 * hi + 1];
            acc = __builtin_amdgcn_wmma_f32_16x16x4_f32(
                false, a, false, bfrag[c], (short)0, acc, false, false);
        }

        // gather all 16 sims per lane (each half holds 8, swap via shfl_xor 16)
        float s[16];
        #pragma unroll
        for (int j = 0; j < 8; ++j) {
            const float own = acc[j] * 0.125f;          // / sqrt(64)
            const float oth = __shfl_xor(own, 16);
            s[j]     = hi ? oth : own;                  // row j
            s[8 + j] = hi ? own : oth;                  // row 8+j
        }

        float tmax = s[0];
        #pragma unroll
        for (int m = 1; m < 16; ++m) tmax = fmaxf(tmax, s[m]);

        const float nm = fmaxf(mW, tmax);
        const float rescale = __expf(mW - nm);          // 0 on first tile
        sW *= rescale; px *= rescale; py *= rescale;

        #pragma unroll
        for (int m = 0; m < 16; ++m) {
            const float e = __expf(s[m] - nm);
            sW += e;
            const float* rp = rowbase + m * HD + 2 * lane;
            px = fmaf(e, rp[0], px);
            py = fmaf(e, rp[1], py);
        }
        mW = nm;
    }

    // ---- merge the 8 waves ----
    if (lane == 0) { lds_ms[wave][0] = mW; lds_ms[wave][1] = sW; }
    lds_pool[wave][2 * lane]     = px;
    lds_pool[wave][2 * lane + 1] = py;
    __syncthreads();

    if (tid < HD) {
        float M = -__builtin_inff();
        #pragma unroll
        for (int w = 0; w < WAVES; ++w) M = fmaxf(M, lds_ms[w][0]);
        float S = 0.f, P = 0.f;
        #pragma unroll
        for (int w = 0; w < WAVES; ++w) {
            const float f = __expf(lds_ms[w][0] - M);
            S += f * lds_ms[w][1];
            P  = fmaf(f, lds_pool[w][tid], P);
        }
        lds_pooled[tid] = P / S;
    }
    __syncthreads();

    // ---- out[b] = pooled @ oW^T + ob ----
    if (tid < HD) {
        const float* wr = oW + tid * HD;
        float acc = ob[tid];
        #pragma unroll 8
        for (int k = 0; k < HD; ++k) acc = fmaf(lds_pooled[k], wr[k], acc);
        out[(size_t)b * HD + tid] = acc;
    }
}

extern "C" void kernel_launch(void* const* d_in, const int* in_sizes, int n_in,
                              void* d_out, int out_size, void* d_ws, size_t ws_size,
                              hipStream_t stream) {
    const int*   query = (const int*)d_in[0];
    const float* mem_  = (const float*)d_in[1];
    const float* emb   = (const float*)d_in[2];
    const float* qW    = (const float*)d_in[3];
    const float* qb    = (const float*)d_in[4];
    const float* oW    = (const float*)d_in[5];
    const float* ob    = (const float*)d_in[6];
    float* out = (float*)d_out;
    const int Bn = in_sizes[0];   // 2048
    attn_pool_kernel<<<Bn, BLOCK, 0, stream>>>(query, mem_, emb, qW, qb, oW, ob, out);
}